// Sent_Posit_Drmm_Modeler_19636590478010
// MI455X (gfx1250) — compile-verified
//
#include <hip/hip_runtime.h>
#include <math.h>

// ---------------- Problem constants ----------------
#define D_EMB   300
#define NF      256
#define FS      3
#define KW      (FS * D_EMB)     // 900, im2col K
#define TOPK    5
#define N_SENTS 200
#define LS      64
#define LS_PAD  66               // 64 + 2 zero rows (conv window tail)
#define LQ      30
#define LQ_SEG  32               // q output rows computed (30 real + 2 zero)
#define LQ_PAD  34               // 32 + 2 zero rows for conv window tail
#define NS_TOT  (N_SENTS * LS)   // 12800
#define SROWS_PAD (N_SENTS * LS_PAD) // 13200 padded sentence rows
#define LDC     NS_TOT           // sim-GEMM leading dim

typedef __attribute__((ext_vector_type(2))) float v2f;
typedef __attribute__((ext_vector_type(8))) float v8f;

// =====================================================================
// Kernel 1: embedding gather into ZERO-PADDED layout + row L2 norms.
// s_emb: [200][66][300] (rows 64,65 of each segment zeroed)
// q_emb: [34][300]      (rows 30..33 zeroed)
// One wave per padded row; pad rows just write zeros.
// =====================================================================
__global__ void gather_norm_kernel(const float* __restrict__ emb,
                                   const int* __restrict__ q_ids,
                                   const int* __restrict__ s_ids,
                                   float* __restrict__ q_emb,
                                   float* __restrict__ s_emb,
                                   float* __restrict__ qn,      // [30]
                                   float* __restrict__ sn)      // [12800]
{
    const int row  = blockIdx.x;
    const int lane = threadIdx.x;
    int tok;
    float* dst;
    float* ndst;
    if (row < SROWS_PAD) {
        const int seg = row / LS_PAD;
        const int l   = row - seg * LS_PAD;
        dst = s_emb + (size_t)row * D_EMB;
        if (l >= LS) {   // zero-pad row
            for (int d = lane; d < D_EMB; d += 32) dst[d] = 0.f;
            return;
        }
        tok  = s_ids[seg * LS + l];
        ndst = sn + seg * LS + l;
    } else {
        const int q = row - SROWS_PAD;
        dst = q_emb + (size_t)q * D_EMB;
        if (q >= LQ) {   // zero-pad row
            for (int d = lane; d < D_EMB; d += 32) dst[d] = 0.f;
            return;
        }
        tok  = q_ids[q];
        ndst = qn + q;
    }
    const float* src = emb + (size_t)tok * D_EMB;
    float ss = 0.f;
    for (int d = lane; d < D_EMB; d += 32) {
        const float v = src[d];
        dst[d] = v;
        ss += v * v;
    }
    #pragma unroll
    for (int off = 16; off > 0; off >>= 1)
        ss += __shfl_down(ss, off, 32);
    if (lane == 0) *ndst = sqrtf(ss);
}

// =====================================================================
// Kernel 2: im2col conv as WMMA f32 GEMM over zero-padded input.
//   C[m, f] = sum_{kk<900} X[(seg*Lpad + l)*300 + kk] * W[f*900 + kk]
// Padding rows make the patch a plain contiguous 900-float window:
// inner loop = 5x global_load_b64 + 4x v_wmma_f32_16x16x4_f32, branch-free.
// Block = 128 threads = 4 waves, one M-tile (16 rows) per wave;
// each wave covers 64 filters (4 N-subtiles, A-frag reused 4x).
// grid = (ceil(Mtiles/4), NF/64)
// =====================================================================
__global__ void conv_gemm_wmma_kernel(const float* __restrict__ X, // [nSeg][Lpad][300]
                                      const int L, const int Lpad, const int Mrows,
                                      const float* __restrict__ W, // [256][900]
                                      float* __restrict__ Cout)    // [Mrows][256]
{
    const int wave = threadIdx.x >> 5;
    const int lane = threadIdx.x & 31;
    const int half = lane >> 4;      // 0: K+0/K+1, 1: K+2/K+3
    const int l15  = lane & 15;

    const int tileM = blockIdx.x * 4 + wave;
    if (tileM * 16 >= Mrows) return;
    const int m0 = tileM * 16;
    const int n0 = blockIdx.y * 64;

    const int m   = m0 + l15;
    const int seg = m / L;
    const int l   = m - seg * L;
    const float* Arow = X + ((size_t)seg * Lpad + l) * D_EMB + half * 2;

    const float* Wr0 = W + (size_t)(n0 + l15)      * KW + half * 2;
    const float* Wr1 = W + (size_t)(n0 + l15 + 16) * KW + half * 2;
    const float* Wr2 = W + (size_t)(n0 + l15 + 32) * KW + half * 2;
    const float* Wr3 = W + (size_t)(n0 + l15 + 48) * KW + half * 2;

    v8f acc0 = {}, acc1 = {}, acc2 = {}, acc3 = {};

    for (int kb = 0; kb < KW; kb += 4) {
        const v2f A  = *(const v2f*)(Arow + kb);
        const v2f B0 = *(const v2f*)(Wr0 + kb);
        const v2f B1 = *(const v2f*)(Wr1 + kb);
        const v2f B2 = *(const v2f*)(Wr2 + kb);
        const v2f B3 = *(const v2f*)(Wr3 + kb);

        acc0 = __builtin_amdgcn_wmma_f32_16x16x4_f32(false, A, false, B0, (short)0, acc0, false, false);
        acc1 = __builtin_amdgcn_wmma_f32_16x16x4_f32(false, A, false, B1, (short)0, acc1, false, false);
        acc2 = __builtin_amdgcn_wmma_f32_16x16x4_f32(false, A, false, B2, (short)0, acc2, false, false);
        acc3 = __builtin_amdgcn_wmma_f32_16x16x4_f32(false, A, false, B3, (short)0, acc3, false, false);
    }

    // C/D layout: VGPR j -> row m0 + j + half*8, col n0 + l15 (+16t)
    #pragma unroll
    for (int j = 0; j < 8; ++j) {
        const int row = m0 + j + half * 8;
        float* crow = Cout + (size_t)row * NF + n0 + l15;
        crow[0]  = acc0[j];
        crow[16] = acc1[j];
        crow[32] = acc2[j];
        crow[48] = acc3[j];
    }
}

// =====================================================================
// Kernel 3: row norms of conv outputs (256-dim). One wave per row.
// =====================================================================
__global__ void conv_norm_kernel(const float* __restrict__ s_conv,
                                 const float* __restrict__ q_conv,
                                 float* __restrict__ sn,
                                 float* __restrict__ qn)
{
    const int row  = blockIdx.x;
    const int lane = threadIdx.x;
    const float* src;
    float* dst;
    if (row < NS_TOT) { src = s_conv + (size_t)row * NF; dst = sn + row; }
    else              { src = q_conv + (size_t)(row - NS_TOT) * NF; dst = qn + (row - NS_TOT); }
    float ss = 0.f;
    #pragma unroll
    for (int i = 0; i < NF / 32; ++i) {
        const float v = src[lane + 32 * i];
        ss += v * v;
    }
    #pragma unroll
    for (int off = 16; off > 0; off >>= 1)
        ss += __shfl_down(ss, off, 32);
    if (lane == 0) *dst = sqrtf(ss);
}

// =====================================================================
// Kernel 4: cosine-numerator GEMM  C[q, ns] = sum_k Q[q,k] * S[row(ns),k]
// Q is zero-padded to 32 rows -> no masking anywhere.
// S row mapping handles the padded s_emb layout: row(n) = (n/Lb)*Lbpad + n%Lb.
// Block = 128 threads = 4 waves; wave w handles N-group (by*4+w)*64.
// grid = (2, 12800/256)
// =====================================================================
__global__ void sim_gemm_wmma_kernel(const float* __restrict__ Q, // [32+][K], rows>=30 zero
                                     const int K,
                                     const float* __restrict__ S,
                                     const int Lb, const int Lbpad,
                                     float* __restrict__ C)       // [32][12800]
{
    const int wave = threadIdx.x >> 5;
    const int lane = threadIdx.x & 31;
    const int half = lane >> 4;
    const int l15  = lane & 15;
    const int m0   = blockIdx.x * 16;
    const int n0   = (blockIdx.y * 4 + wave) * 64;

    const float* Arow = Q + (size_t)(m0 + l15) * K + half * 2;

    const int nA = n0 + l15;
    const int nB = n0 + l15 + 16;
    const int nCc = n0 + l15 + 32;
    const int nD = n0 + l15 + 48;
    const float* Sr0 = S + ((size_t)(nA / Lb) * Lbpad + (nA % Lb)) * K + half * 2;
    const float* Sr1 = S + ((size_t)(nB / Lb) * Lbpad + (nB % Lb)) * K + half * 2;
    const float* Sr2 = S + ((size_t)(nCc / Lb) * Lbpad + (nCc % Lb)) * K + half * 2;
    const float* Sr3 = S + ((size_t)(nD / Lb) * Lbpad + (nD % Lb)) * K + half * 2;

    v8f acc0 = {}, acc1 = {}, acc2 = {}, acc3 = {};

    for (int kb = 0; kb < K; kb += 4) {
        const v2f A  = *(const v2f*)(Arow + kb);
        const v2f B0 = *(const v2f*)(Sr0 + kb);
        const v2f B1 = *(const v2f*)(Sr1 + kb);
        const v2f B2 = *(const v2f*)(Sr2 + kb);
        const v2f B3 = *(const v2f*)(Sr3 + kb);

        acc0 = __builtin_amdgcn_wmma_f32_16x16x4_f32(false, A, false, B0, (short)0, acc0, false, false);
        acc1 = __builtin_amdgcn_wmma_f32_16x16x4_f32(false, A, false, B1, (short)0, acc1, false, false);
        acc2 = __builtin_amdgcn_wmma_f32_16x16x4_f32(false, A, false, B2, (short)0, acc2, false, false);
        acc3 = __builtin_amdgcn_wmma_f32_16x16x4_f32(false, A, false, B3, (short)0, acc3, false, false);
    }

    #pragma unroll
    for (int j = 0; j < 8; ++j) {
        const int row = m0 + j + half * 8;
        float* crow = C + (size_t)row * LDC + n0 + l15;
        crow[0]  = acc0[j];
        crow[16] = acc1[j];
        crow[32] = acc2[j];
        crow[48] = acc3[j];
    }
}

// =====================================================================
// Kernel 5: fused normalize + mask + top-5 pool + logistic.
// One thread per (n, q). 3 channels x 64 values, branch-free top-5.
// =====================================================================
__device__ __forceinline__ void ins5(float t[TOPK], float v)
{
    #pragma unroll
    for (int i = 0; i < TOPK; ++i) {
        const float hi = fmaxf(t[i], v);
        const float lo = fminf(t[i], v);
        t[i] = hi;
        v = lo;
    }
}

__global__ void pool_logit_kernel(const float* __restrict__ num_ins,  // [32][12800]
                                  const float* __restrict__ num_sens, // [32][12800]
                                  const float* __restrict__ qn_ins,
                                  const float* __restrict__ sn_ins,
                                  const float* __restrict__ qn_sens,
                                  const float* __restrict__ sn_sens,
                                  const int* __restrict__ question,
                                  const int* __restrict__ sents,     // [12800]
                                  const float* __restrict__ dsim,    // [200][64][30]
                                  const float* __restrict__ lin_w,   // [6]
                                  const float* __restrict__ lin_b,   // [1]
                                  float* __restrict__ lo)            // [200*30]
{
    const int t = blockIdx.x * blockDim.x + threadIdx.x;
    if (t >= N_SENTS * LQ) return;
    const int n = t / LQ;
    const int q = t - n * LQ;

    const bool  qm  = question[q] > 1;
    const float qni = qn_ins[q];
    const float qns = qn_sens[q];

    float ta[TOPK], tb[TOPK], tc[TOPK];
    #pragma unroll
    for (int i = 0; i < TOPK; ++i) { ta[i] = -1e30f; tb[i] = -1e30f; tc[i] = -1e30f; }

    for (int s = 0; s < LS; ++s) {
        const int ns = n * LS + s;
        const float numi = num_ins[(size_t)q * LDC + ns];
        float v1 = 0.f;
        if (qm && sents[ns] > 1) v1 = numi / (qni * sn_ins[ns]);
        const float v2 = num_sens[(size_t)q * LDC + ns] / (qns * sn_sens[ns]);
        const float v3 = dsim[(size_t)ns * LQ + q];
        ins5(ta, v1);
        ins5(tb, v2);
        ins5(tc, v3);
    }

    const float f0 = ta[0];
    const float f1 = (ta[0] + ta[1] + ta[2] + ta[3] + ta[4]) * 0.2f;
    const float f2 = tb[0];
    const float f3 = (tb[0] + tb[1] + tb[2] + tb[3] + tb[4]) * 0.2f;
    const float f4 = tc[0];
    const float f5 = (tc[0] + tc[1] + tc[2] + tc[3] + tc[4]) * 0.2f;

    const float z = f0 * lin_w[0] + f1 * lin_w[1] + f2 * lin_w[2] +
                    f3 * lin_w[3] + f4 * lin_w[4] + f5 * lin_w[5] + lin_b[0];
    lo[t] = 1.f / (1.f + expf(-z));
}

// =====================================================================
// Kernel 6: sent_out[n] = mean_q lo[n,q]; doc_emit = mean_n sent_out.
// d_out[0] = doc_emit, d_out[1..200] = sent_out. Single block.
// =====================================================================
__global__ void finalize_kernel(const float* __restrict__ lo, float* __restrict__ out)
{
    __shared__ float sm[256];
    const int t = threadIdx.x;
    float s = 0.f;
    if (t < N_SENTS) {
        #pragma unroll
        for (int q = 0; q < LQ; ++q) s += lo[t * LQ + q];
        s *= (1.f / (float)LQ);
        out[1 + t] = s;
    }
    sm[t] = (t < N_SENTS) ? s : 0.f;
    __syncthreads();
    #pragma unroll
    for (int off = 128; off > 0; off >>= 1) {
        if (t < off) sm[t] += sm[t + off];
        __syncthreads();
    }
    if (t == 0) out[0] = sm[0] * (1.f / (float)N_SENTS);
}

// =====================================================================
// Host launcher
// =====================================================================
extern "C" void kernel_launch(void* const* d_in, const int* in_sizes, int n_in,
                              void* d_out, int out_size, void* d_ws, size_t ws_size,
                              hipStream_t stream)
{
    const float* emb      = (const float*)d_in[0];  // [400000][300]
    const float* filt     = (const float*)d_in[1];  // [256][1][3][300]
    const float* lin_w    = (const float*)d_in[2];  // [1][6]
    const float* lin_b    = (const float*)d_in[3];  // [1]
    const int*   question = (const int*)  d_in[4];  // [30]
    const int*   sents    = (const int*)  d_in[5];  // [200][64]
    const float* dsim     = (const float*)d_in[6];  // [200][64][30]
    float*       out      = (float*)d_out;          // [201]

    float* ws = (float*)d_ws;
    size_t o = 0;
    float* q_emb    = ws + o; o += (size_t)LQ_PAD * D_EMB;      // [34][300]
    float* s_emb    = ws + o; o += (size_t)SROWS_PAD * D_EMB;   // [200][66][300]
    float* qn_ins   = ws + o; o += 32;
    float* sn_ins   = ws + o; o += NS_TOT;
    float* q_conv   = ws + o; o += (size_t)32 * NF;             // rows 30,31 == 0
    float* s_conv   = ws + o; o += (size_t)NS_TOT * NF;
    float* qn_sens  = ws + o; o += 32;
    float* sn_sens  = ws + o; o += NS_TOT;
    float* num_ins  = ws + o; o += (size_t)32 * LDC;
    float* num_sens = ws + o; o += (size_t)32 * LDC;
    float* lo       = ws + o; o += (size_t)N_SENTS * LQ;

    // 1) gather embeddings into zero-padded layout + 300-dim norms
    gather_norm_kernel<<<SROWS_PAD + LQ_PAD, 32, 0, stream>>>(
        emb, question, sents, q_emb, s_emb, qn_ins, sn_ins);

    // 2) conv as branch-free im2col WMMA GEMM
    //    sentences: M=12800 (800 tiles, 4 per block), query: M=32 (2 tiles)
    conv_gemm_wmma_kernel<<<dim3(NS_TOT / 64, NF / 64), 128, 0, stream>>>(
        s_emb, LS, LS_PAD, NS_TOT, filt, s_conv);
    conv_gemm_wmma_kernel<<<dim3(1, NF / 64), 128, 0, stream>>>(
        q_emb, LQ_SEG, LQ_PAD, LQ_SEG, filt, q_conv);

    // 3) 256-dim norms of conv outputs
    conv_norm_kernel<<<NS_TOT + LQ, 32, 0, stream>>>(s_conv, q_conv, sn_sens, qn_sens);

    // 4) cosine numerators as WMMA GEMMs: [32 x 12800]
    sim_gemm_wmma_kernel<<<dim3(2, NS_TOT / 256), 128, 0, stream>>>(
        q_emb, D_EMB, s_emb, LS, LS_PAD, num_ins);
    sim_gemm_wmma_kernel<<<dim3(2, NS_TOT / 256), 128, 0, stream>>>(
        q_conv, NF, s_conv, NS_TOT, NS_TOT, num_sens);

    // 5) fused normalize + mask + top-5 pool + logistic
    pool_logit_kernel<<<(N_SENTS * LQ + 127) / 128, 128, 0, stream>>>(
        num_ins, num_sens, qn_ins, sn_ins, qn_sens, sn_sens,
        question, sents, dsim, lin_w, lin_b, lo);

    // 6) means -> d_out
    finalize_kernel<<<1, 256, 0, stream>>>(lo, out);
}